// FullAttention_68590627717416
// MI455X (gfx1250) — compile-verified
//
#include <hip/hip_runtime.h>

#define BB 4
#define LL 2048
#define HH 8
#define EE 64
#define DD 64

typedef float v2f __attribute__((ext_vector_type(2)));
typedef float v8f __attribute__((ext_vector_type(8)));

__device__ __forceinline__ v8f wmma_f32(v2f a, v2f b, v8f c) {
    // D = A(16x4) * B(4x16) + C, full fp32
    return __builtin_amdgcn_wmma_f32_16x16x4_f32(
        /*neg_a=*/false, a, /*neg_b=*/false, b,
        /*c_mod=*/(short)0, c, /*reuse_a=*/false, /*reuse_b=*/false);
}

__global__ __launch_bounds__(128)
void attn_flash_f32wmma(const float* __restrict__ Q,
                        const float* __restrict__ K,
                        const float* __restrict__ V,
                        const float* __restrict__ Sg,
                        float* __restrict__ Out)
{
    const int t  = threadIdx.x & 31;   // lane in wave32
    const int wv = threadIdx.x >> 5;   // wave in block (0..3)
    const int m  = t & 15;             // query column / fragment row
    const int hi = t >> 4;             // lane half

    const int mt = blockIdx.x % (LL / 64);
    const int bh = blockIdx.x / (LL / 64);
    const int b  = bh / HH;
    const int h  = bh % HH;
    const int l0 = mt * 64 + wv * 16;  // first query row of this wave
    const int l  = l0 + m;             // this lane's query row

    const float* qrow  = Q + (((size_t)b * LL + l) * HH + h) * EE;
    const float* kbase = K + (((size_t)b * LL) * HH + h) * EE;  // + s*HH*EE
    const float* vbase = V + (((size_t)b * LL) * HH + h) * DD;  // + s*HH*DD

    // ---- Q fragments: B-matrix of S^T = K * Q^T. Lane t supplies Q[m][4kk+2hi .. +1]
    v2f qf[16];
    #pragma unroll
    for (int kk = 0; kk < 16; ++kk)
        qf[kk] = *(const v2f*)(qrow + 4 * kk + 2 * hi);

    // ---- per-query Gaussian bias constants (per-lane scalars)
    float sv    = Sg[((size_t)b * HH + h) * LL + l];
    float sg    = 1.0f / (1.0f + __expf(-5.0f * sv)) + 1e-5f;   // sigmoid(5x)+1e-5
    float sp    = __expf(sg * 1.0986122886681098f) - 1.0f;      // 3^sg - 1
    float cl    = 0.3989422804014327f / sp;                     // 1/(sqrt(2pi)*sig)
    float ninv2 = -0.5f / (sp * sp);                            // -1/(2 sig^2)

    const float scale = 0.125f;  // 1/sqrt(E)

    float runM = -__builtin_inff();
    float runS = 0.0f;
    v8f o0 = {}, o1 = {}, o2 = {}, o3 = {};   // O^T chunks: o_c[v]=O^T[16c+8hi+v][m]

    for (int n0 = 0; n0 <= l0; n0 += 16) {
        // ---- S^T tile = K_tile(16x64) * Q^T(64x16)
        // two independent 8-deep WMMA chains (even/odd k-step) for ILP
        v8f st0 = {}, st1 = {};
        const float* krow = kbase + (size_t)(n0 + m) * (HH * EE);
        #pragma unroll
        for (int kk = 0; kk < 8; ++kk) {
            v2f kfa = *(const v2f*)(krow + 8 * kk + 2 * hi);
            v2f kfb = *(const v2f*)(krow + 8 * kk + 4 + 2 * hi);
            st0 = wmma_f32(kfa, qf[2 * kk],     st0);
            st1 = wmma_f32(kfb, qf[2 * kk + 1], st1);
        }
        v8f st = st0 + st1;

        // ---- V^T fragments (A-matrix of O^T += V^T * P)
        // lane t: {V[n0+4kk+2hi][16c+m], V[n0+4kk+2hi+1][16c+m]}
        v2f vf[4][4];
        #pragma unroll
        for (int kk = 0; kk < 4; ++kk) {
            const float* vr0 = vbase + (size_t)(n0 + 4 * kk + 2 * hi) * (HH * DD) + m;
            const float* vr1 = vr0 + HH * DD;
            #pragma unroll
            for (int c = 0; c < 4; ++c) {
                vf[c][kk].x = vr0[16 * c];
                vf[c][kk].y = vr1[16 * c];
            }
        }

        // ---- logits: st[v] = S^T[8hi+v][m], key s = n0+8hi+v, query l
        float lg[8];
        float tmax = -__builtin_inff();
        #pragma unroll
        for (int v = 0; v < 8; ++v) {
            int   s    = n0 + 8 * hi + v;
            float ss   = (float)(s * s);
            float bias = cl * __expf(ss * ninv2);
            float e    = scale * (st[v] + bias);
            if (s > l) e = -__builtin_inff();     // causal mask (only hits diag tile)
            lg[v] = e;
            tmax  = fmaxf(tmax, e);
        }
        tmax = fmaxf(tmax, __shfl_xor(tmax, 16, 32));   // combine halves (same query m)

        float newM  = fmaxf(runM, tmax);
        float alpha = __expf(runM - newM);

        float pexp[8];
        float ts = 0.0f;
        #pragma unroll
        for (int v = 0; v < 8; ++v) {
            pexp[v] = __expf(lg[v] - newM);
            ts += pexp[v];
        }
        ts  += __shfl_xor(ts, 16, 32);
        runS = runS * alpha + ts;
        runM = newM;

        o0 *= alpha; o1 *= alpha; o2 *= alpha; o3 *= alpha;

        // ---- build B-fragments of P (4x16 slices): lane t needs P[4kk+2hi+j][m]
        // own regs cover rows 8hi+v; grab the other half's rows with shfl_xor(16)
        float poth[8];
        #pragma unroll
        for (int v = 0; v < 8; ++v)
            poth[v] = __shfl_xor(pexp[v], 16, 32);

        v2f pb[4];
        pb[0].x = hi ? poth[2] : pexp[0];  pb[0].y = hi ? poth[3] : pexp[1];
        pb[1].x = hi ? poth[6] : pexp[4];  pb[1].y = hi ? poth[7] : pexp[5];
        pb[2].x = hi ? pexp[2] : poth[0];  pb[2].y = hi ? pexp[3] : poth[1];
        pb[3].x = hi ? pexp[6] : poth[4];  pb[3].y = hi ? pexp[7] : poth[5];

        // ---- O^T += V^T * P : 4 independent chains (d-chunks) x 4 K-steps
        #pragma unroll
        for (int kk = 0; kk < 4; ++kk) {
            o0 = wmma_f32(vf[0][kk], pb[kk], o0);
            o1 = wmma_f32(vf[1][kk], pb[kk], o1);
            o2 = wmma_f32(vf[2][kk], pb[kk], o2);
            o3 = wmma_f32(vf[3][kk], pb[kk], o3);
        }
    }

    // ---- normalize and store: Out[b][l][h][d], d = 16c + 8hi + v (8 consecutive)
    float inv = 1.0f / runS;
    float* orow = Out + (((size_t)b * LL + l) * HH + h) * DD + 8 * hi;
    #pragma unroll
    for (int v = 0; v < 8; ++v) {
        orow[v]      = o0[v] * inv;
        orow[16 + v] = o1[v] * inv;
        orow[32 + v] = o2[v] * inv;
        orow[48 + v] = o3[v] * inv;
    }
}

extern "C" void kernel_launch(void* const* d_in, const int* in_sizes, int n_in,
                              void* d_out, int out_size, void* d_ws, size_t ws_size,
                              hipStream_t stream) {
    const float* q  = (const float*)d_in[0];
    const float* k  = (const float*)d_in[1];
    const float* v  = (const float*)d_in[2];
    const float* sg = (const float*)d_in[3];
    // d_in[4] is the triangular causal mask; it is reproduced analytically.
    (void)in_sizes; (void)n_in; (void)d_ws; (void)ws_size; (void)out_size;

    dim3 grid(BB * HH * (LL / 64));
    attn_flash_f32wmma<<<grid, 128, 0, stream>>>(q, k, v, sg, (float*)d_out);
}